// Up2Down_layer_82669530513964
// MI455X (gfx1250) — compile-verified
//
#include <hip/hip_runtime.h>
#include <hip/hip_bf16.h>

typedef __attribute__((ext_vector_type(2))) float v2f;
typedef __attribute__((ext_vector_type(4))) float v4f;
typedef __attribute__((ext_vector_type(8))) float v8f;

#define FEAT 128
#define A_STRIDE 132  // padded LDS stride (floats) to avoid bank conflicts

// ---------------------------------------------------------------------------
// Pass 1a: deg[n] = 1 (self-loop contribution)
__global__ void deg_init_kernel(float* __restrict__ deg, int n) {
    int i = blockIdx.x * blockDim.x + threadIdx.x;
    if (i < n) deg[i] = 1.0f;
}

// Pass 1b: deg[dst[e]] += 1
__global__ void deg_count_kernel(const int* __restrict__ dst,
                                 float* __restrict__ deg, int e) {
    int i = blockIdx.x * blockDim.x + threadIdx.x;
    if (i < e) unsafeAtomicAdd(&deg[dst[i]], 1.0f);
}

// Pass 1c: deg -> deg^{-1/2}  (deg >= 1 always, so rsqrt is safe)
__global__ void dinv_kernel(float* __restrict__ deg, int n) {
    int i = blockIdx.x * blockDim.x + threadIdx.x;
    if (i < n) deg[i] = rsqrtf(deg[i]);
}

// ---------------------------------------------------------------------------
// Pass 2: initialize out with the self-loop term: out[n,:] = emb[n,:] / deg[n]
// (norm for a self loop is dinv[n]^2). One thread per 4 features (float4).
__global__ void self_loop_kernel(const float* __restrict__ emb,
                                 const float* __restrict__ dinv,
                                 float* __restrict__ out, int n) {
    int t = blockIdx.x * blockDim.x + threadIdx.x;
    int node = t >> 5;          // 32 threads per node (128 feats / 4)
    int c = (t & 31) * 4;
    if (node >= n) return;
    float s = dinv[node];
    s = s * s;
    v4f v = *(const v4f*)(emb + (size_t)node * FEAT + c);
    v *= s;
    *(v4f*)(out + (size_t)node * FEAT + c) = v;
}

// ---------------------------------------------------------------------------
// Pass 3: edge scatter: out[dst,:] += norm(e) * emb[src,:]
// One wave32 per edge; each lane handles 4 contiguous floats.
__global__ void scatter_kernel(const float* __restrict__ emb,
                               const int* __restrict__ src,
                               const int* __restrict__ dst,
                               const float* __restrict__ dinv,
                               float* __restrict__ out, int e) {
    int gt = blockIdx.x * blockDim.x + threadIdx.x;
    int edge = gt >> 5;
    int c = (gt & 31) * 4;
    if (edge >= e) return;
    int s = src[edge];
    int d = dst[edge];
    float norm = dinv[s] * dinv[d];
    v4f v = *(const v4f*)(emb + (size_t)s * FEAT + c);
    float* o = out + (size_t)d * FEAT + c;
    unsafeAtomicAdd(o + 0, v.x * norm);
    unsafeAtomicAdd(o + 1, v.y * norm);
    unsafeAtomicAdd(o + 2, v.z * norm);
    unsafeAtomicAdd(o + 3, v.w * norm);
}

// ---------------------------------------------------------------------------
// Pass 4: in-place row-local GEMM: out[16 rows, :] = out[16 rows, :] @ W + b
// Block = 256 threads = 8 waves. Wave w computes output cols [16w, 16w+16)
// via V_WMMA_F32_16X16X4_F32, K-loop of 32 steps over the 128-deep K dim.
// Two accumulators break the WMMA->WMMA RAW chain; the block's 16x128 input
// tile is staged in LDS behind a barrier so the in-place overwrite cannot
// race the reads.
__global__ __launch_bounds__(256) void gcn_gemm_kernel(
    const float* __restrict__ Wmat, const float* __restrict__ bias,
    float* __restrict__ out, int n) {
    __shared__ float Ain[16 * A_STRIDE];

    const int tid = threadIdx.x;
    const int row0 = blockIdx.x * 16;
    const bool full_tile = (row0 + 16 <= n);   // uniform across the block

    // Stage 16x128 floats (2048) with 256 threads: 2 float4 each.
    if (full_tile) {
        #pragma unroll
        for (int j = 0; j < 2; ++j) {
            int i = tid * 2 + j;            // 0..511
            int r = i >> 5;                 // row within tile
            int c4 = i & 31;                // float4 column index
            v4f v = *(const v4f*)(out + (size_t)(row0 + r) * FEAT + c4 * 4);
            *(v4f*)(&Ain[r * A_STRIDE + c4 * 4]) = v;
        }
    } else {
        #pragma unroll
        for (int j = 0; j < 2; ++j) {
            int i = tid * 2 + j;
            int r = i >> 5;
            int c4 = i & 31;
            int row = row0 + r;
            v4f v = {};
            if (row < n) v = *(const v4f*)(out + (size_t)row * FEAT + c4 * 4);
            *(v4f*)(&Ain[r * A_STRIDE + c4 * 4]) = v;
        }
    }
    __syncthreads();

    const int wave = tid >> 5;          // 0..7 -> N tile
    const int lane = tid & 31;
    const int l = lane & 15;
    const int h = lane >> 4;
    const int col = wave * 16 + l;      // global output column

    v8f acc0 = {};
    v8f acc1 = {};
    #pragma unroll 8
    for (int k = 0; k < FEAT / 4; ++k) {
        const int kk = 4 * k + 2 * h;
        v2f a, b;
        // A fragment (16x4 f32): lane l(+16h) holds A[l][2h], A[l][2h+1]
        a.x = Ain[l * A_STRIDE + kk];
        a.y = Ain[l * A_STRIDE + kk + 1];
        // B fragment (4x16 f32): lane l(+16h) holds B[2h][l], B[2h+1][l]
        b.x = Wmat[(size_t)kk * FEAT + col];
        b.y = Wmat[(size_t)(kk + 1) * FEAT + col];
        if (k & 1) {
            acc1 = __builtin_amdgcn_wmma_f32_16x16x4_f32(
                false, a, false, b, (short)0, acc1, false, false);
        } else {
            acc0 = __builtin_amdgcn_wmma_f32_16x16x4_f32(
                false, a, false, b, (short)0, acc0, false, false);
        }
    }
    v8f acc = acc0 + acc1;

    const float bv = bias[col];
    if (full_tile) {
        float* orow = out + (size_t)(row0 + 8 * h) * FEAT + col;
        #pragma unroll
        for (int i = 0; i < 8; ++i)
            orow[(size_t)i * FEAT] = acc[i] + bv;   // C/D: VGPR i = row i+8h
    } else {
        #pragma unroll
        for (int i = 0; i < 8; ++i) {
            int row = row0 + i + 8 * h;
            if (row < n) out[(size_t)row * FEAT + col] = acc[i] + bv;
        }
    }
}

// ---------------------------------------------------------------------------
extern "C" void kernel_launch(void* const* d_in, const int* in_sizes, int n_in,
                              void* d_out, int out_size, void* d_ws, size_t ws_size,
                              hipStream_t stream) {
    const float* emb  = (const float*)d_in[0];
    const int*   eidx = (const int*)d_in[1];
    const float* Wmat = (const float*)d_in[2];
    const float* bias = (const float*)d_in[3];
    float* out = (float*)d_out;

    const int N = in_sizes[0] / FEAT;   // 200000
    const int E = in_sizes[1] / 2;      // 600000
    const int* src = eidx;              // edge_index[0]
    const int* dst = eidx + E;          // edge_index[1]

    float* dinv = (float*)d_ws;         // N floats of workspace (deg -> dinv)

    const int T = 256;
    deg_init_kernel <<<(N + T - 1) / T, T, 0, stream>>>(dinv, N);
    deg_count_kernel<<<(E + T - 1) / T, T, 0, stream>>>(dst, dinv, E);
    dinv_kernel     <<<(N + T - 1) / T, T, 0, stream>>>(dinv, N);
    self_loop_kernel<<<(N * 32 + T - 1) / T, T, 0, stream>>>(emb, dinv, out, N);
    scatter_kernel  <<<(E * 32 + T - 1) / T, T, 0, stream>>>(emb, src, dst, dinv, out, E);
    gcn_gemm_kernel <<<(N + 15) / 16, T, 0, stream>>>(Wmat, bias, out, N);
}